// CoralLoss_13640816132487
// MI455X (gfx1250) — compile-verified
//
#include <hip/hip_runtime.h>
#include <hip/hip_bf16.h>

// ---------------------------------------------------------------------------
// CORAL ordinal loss, mean-reduced.  B=262144 rows, KM1=100 thresholds.
// Memory-bound streaming reduction: ~105MB logits @ 23.3 TB/s -> ~4.5us floor.
// CDNA5 path: async global->LDS (ASYNCcnt) double-buffered streaming.
// ---------------------------------------------------------------------------

#define TPB       256                 // 8 waves (wave32)
#define VPT       4                   // float4 per thread per stage
#define STAGE_V4  (TPB * VPT)         // 1024 float4 = 16KB per stage
#define SPB       8                   // stages per block (128KB contiguous)
#define RV4       25                  // float4 per row (KM1/4 = 100/4)

#if defined(__HIP_DEVICE_COMPILE__) && \
    __has_builtin(__builtin_amdgcn_global_load_async_to_lds_b128)
#define USE_ASYNC_LDS 1
#else
#define USE_ASYNC_LDS 0
#endif

#if USE_ASYNC_LDS
typedef int v4i __attribute__((ext_vector_type(4)));
typedef __attribute__((address_space(1))) v4i as1_v4i;   // global ("__device__")
typedef __attribute__((address_space(3))) v4i as3_v4i;   // LDS ("__shared__")
#endif

template <int N>
__device__ __forceinline__ void wait_asynccnt_le() {
#if defined(__HIP_DEVICE_COMPILE__)
#if __has_builtin(__builtin_amdgcn_s_wait_asynccnt)
  __builtin_amdgcn_s_wait_asynccnt((short)N);
#else
  asm volatile("s_wait_asynccnt %0" ::"n"(N));
#endif
  asm volatile("" ::: "memory");  // keep LDS reads after the wait
#endif
}

// Drain this wave's outstanding DS (LDS) loads before an ASYNC write may
// reuse the same LDS buffer (ASYNC is unordered with DS ops).
__device__ __forceinline__ void wait_dscnt0() {
#if defined(__HIP_DEVICE_COMPILE__)
#if __has_builtin(__builtin_amdgcn_s_wait_dscnt)
  __builtin_amdgcn_s_wait_dscnt((short)0);
#else
  asm volatile("s_wait_dscnt 0");
#endif
  asm volatile("" ::: "memory");
#endif
}

// Numerically-stable BCE-with-logits term (matches jnp formulation).
// max(x,0) - x*level + log1p(exp(-|x|)); hardware exp/log, ~2ulp.
__device__ __forceinline__ float bce_term(float x, bool level) {
  float ax = __builtin_fabsf(x);
  float sp = __logf(1.0f + __expf(-ax));   // v_exp_f32 + v_log_f32
  float mx = fmaxf(x, 0.0f);
  return mx - (level ? x : 0.0f) + sp;
}

__global__ void __launch_bounds__(TPB)
coral_partial_kernel(const float4* __restrict__ logits4,
                     const int*    __restrict__ targets,
                     float*        __restrict__ partials,
                     int tv4) {
#if USE_ASYNC_LDS
  __shared__ float4 tile[2][STAGE_V4];    // 32KB double buffer
#endif
  const int tid    = threadIdx.x;
  const int stage0 = blockIdx.x * SPB;

#if USE_ASYNC_LDS
  auto issue_stage = [&](int s, int buf) {
    const int v0 = (stage0 + s) * STAGE_V4 + tid;
#pragma unroll
    for (int j = 0; j < VPT; ++j) {
      int v  = v0 + j * TPB;
      int vc = (v < tv4) ? v : (tv4 - 1);           // clamp: always valid addr
      __builtin_amdgcn_global_load_async_to_lds_b128(
          (as1_v4i*)(logits4 + vc),
          (as3_v4i*)&tile[buf][tid + j * TPB],
          0, 0);
    }
  };
  issue_stage(0, 0);
  if (SPB > 1) issue_stage(1, 1);
#endif

  float acc = 0.0f;

  for (int s = 0; s < SPB; ++s) {
    const int vbase = (stage0 + s) * STAGE_V4 + tid;

#if USE_ASYNC_LDS
    if (s + 1 < SPB) wait_asynccnt_le<VPT>();   // only next stage in flight
    else             wait_asynccnt_le<0>();     // all landed
#endif

    float4 q[VPT];
#pragma unroll
    for (int j = 0; j < VPT; ++j) {
      int v  = vbase + j * TPB;
      int vc = (v < tv4) ? v : (tv4 - 1);
#if USE_ASYNC_LDS
      (void)vc;
      q[j] = tile[s & 1][tid + j * TPB];
#else
      q[j] = logits4[vc];
      int vp = v + STAGE_V4;                        // software prefetch
      if (vp < tv4) __builtin_prefetch(logits4 + vp, 0, 0);
#endif
    }

#if USE_ASYNC_LDS
    // ds_loads of tile[s&1] must fully drain before async writes reuse it
    // (ASYNCcnt and DScnt pipelines are mutually unordered).
    if (s + 2 < SPB) {
      wait_dscnt0();
      issue_stage(s + 2, s & 1);
    }
#endif

#pragma unroll
    for (int j = 0; j < VPT; ++j) {
      int v  = vbase + j * TPB;
      int vc = (v < tv4) ? v : (tv4 - 1);
      float    w   = (v < tv4) ? 1.0f : 0.0f;
      unsigned uv  = (unsigned)vc;
      unsigned row = uv / RV4;                      // const-div -> mulhi
      int      k0  = (int)(uv % RV4) * 4;
      int      t   = targets[row];                  // L0/L2-resident (1MB)
      float4   x   = q[j];
      float sum = bce_term(x.x, t > k0 + 0) +
                  bce_term(x.y, t > k0 + 1) +
                  bce_term(x.z, t > k0 + 2) +
                  bce_term(x.w, t > k0 + 3);
      acc += w * sum;
    }
  }

  // wave32 butterfly reduction (fixed order -> deterministic)
#pragma unroll
  for (int off = 16; off > 0; off >>= 1)
    acc += __shfl_xor(acc, off, 32);

  __shared__ float wsums[TPB / 32];
  const int wid = tid >> 5, lane = tid & 31;
  if (lane == 0) wsums[wid] = acc;
  __syncthreads();
  if (tid == 0) {
    float sblk = 0.0f;
#pragma unroll
    for (int i = 0; i < TPB / 32; ++i) sblk += wsums[i];
    partials[blockIdx.x] = sblk;
  }
}

__global__ void __launch_bounds__(256)
coral_finalize_kernel(const float* __restrict__ partials, int nb,
                      double inv_n, float* __restrict__ out) {
  __shared__ double sm[256];
  const int tid = threadIdx.x;
  double a = 0.0;
  for (int i = tid; i < nb; i += 256) a += (double)partials[i];
  sm[tid] = a;
  __syncthreads();
  for (int off = 128; off > 0; off >>= 1) {
    if (tid < off) sm[tid] += sm[tid + off];
    __syncthreads();
  }
  if (tid == 0) out[0] = (float)(sm[0] * inv_n);
}

extern "C" void kernel_launch(void* const* d_in, const int* in_sizes, int n_in,
                              void* d_out, int out_size, void* d_ws, size_t ws_size,
                              hipStream_t stream) {
  const float4* logits4 = (const float4*)d_in[0];   // (B, 100) f32, row = 400B (16B aligned)
  const int*    targets = (const int*)d_in[1];      // integer targets per harness convention
  float*        out     = (float*)d_out;            // scalar mean
  float*        partials = (float*)d_ws;

  const int n_logits     = in_sizes[0];             // B * KM1
  const int tv4          = n_logits / 4;            // total float4s
  const int total_stages = (tv4 + STAGE_V4 - 1) / STAGE_V4;
  const int nb           = (total_stages + SPB - 1) / SPB;

  coral_partial_kernel<<<nb, TPB, 0, stream>>>(logits4, targets, partials, tv4);

  const double inv_n = 1.0 / (double)n_logits;
  coral_finalize_kernel<<<1, 256, 0, stream>>>(partials, nb, inv_n, out);
}